// AmortNetwork_70042326663697
// MI455X (gfx1250) — compile-verified
//
#include <hip/hip_runtime.h>
#include <math.h>

// CDNA5 / gfx1250, wave32. Compute-bound (~50 GFLOP, all L2-resident on the
// 192MB L2). f32 WMMA (V_WMMA_F32_16X16X4_F32) since Cholesky needs f32.
// Gram matrices: K-panels of U staged *K-major* in LDS so the WMMA loop is
// 2x ds_load_b64 (immediate offsets) + 1 wmma per K-step.
typedef __attribute__((ext_vector_type(2))) float v2f;
typedef __attribute__((ext_vector_type(8))) float v8f;

#define SAMP 4
#define NROW 2048
#define MIND 256

// ---------------------------------------------------------------------------
// Fused per-(s,d) system solver. One 256-thread workgroup per (s, d).
// LDS: A[LDdim rows x stA]   q_prec -> Lp -> Linv(upper) -> q_cov -> Lc
//      WpT[LDdim x KPp]      prec-scaled K-panel of U, K-major (KPp = KP+2)
// Gram = sum over panels of panel^T panel via WMMA, accumulated in LDS A.
// ---------------------------------------------------------------------------
__global__ __launch_bounds__(256) void gram_chol_sample(
    const float* __restrict__ Upad, int ldU,        // [S, 256, ldU], 0-padded
    const float* __restrict__ plp,                  // [256, Dout]
    const float* __restrict__ pmu,                  // [256, Dout]
    const float* __restrict__ eps,                  // [S, Dout, Din]
    float* __restrict__ wbuf,                       // [S, 256, 272]
    float* __restrict__ klpart,                     // [S*256]
    int Din, int LDdim, int stA, int KP, int Dout)
{
  const int KPp = KP + 2;                     // bank-conflict pad, keeps 8B align
  extern __shared__ float smem[];
  float* A    = smem;                         // LDdim * stA
  float* WpT  = A + (size_t)LDdim * stA;      // LDdim * KPp, K-major
  float* sp   = WpT + (size_t)LDdim * KPp;    // 256  sqrt(prec)
  float* pmv  = sp + 256;                     // 256  prec*pmu
  float* v    = pmv + 256;                    // LDdim  U^T Lambda mu
  float* tv   = v + LDdim;                    // LDdim  Linv * v
  float* qmu  = tv + LDdim;                   // LDdim
  float* dinv = qmu + LDdim;                  // LDdim  1/diag(Lp)
  float* red  = dinv + LDdim;                 // 256

  const int tid = threadIdx.x;
  const int d = blockIdx.x, s = blockIdx.y;
  const int n = Din;
  const float* U = Upad + (size_t)s * MIND * ldU;

  // per-inducing-point scale factors for this output dim d
  if (tid < MIND) {
    float lp = plp[(size_t)tid * Dout + d];
    sp[tid]  = expf(0.5f * lp);
    pmv[tid] = expf(lp) * pmu[(size_t)tid * Dout + d];
  }
  __syncthreads();

  // ---- Gram via WMMA from K-major LDS panels ------------------------------
  {
    const int lane = tid & 31;
    const int wave = tid >> 5;
    const int r    = lane & 15;   // A row / B col / D col within tile
    const int half = lane >> 4;   // K split (K = 2*half, 2*half+1)
    const int nt   = LDdim >> 4;
    const int ntiles = nt * nt;
    for (int kp = 0; kp < MIND; kp += KP) {
      // stage: WpT[col][mm] = U[kp+mm][col] * sp[kp+mm]
      // (reads row-coalesced across threads; each thread owns a WpT row ->
      //  contiguous, conflict-free LDS writes)
      for (int col = tid; col < LDdim; col += 256) {
        float* wrow = WpT + (size_t)col * KPp;
        for (int mm = 0; mm < KP; ++mm)
          wrow[mm] = U[(size_t)(kp + mm) * ldU + col] * sp[kp + mm];
      }
      __syncthreads();
      for (int t = wave; t < ntiles; t += 8) {
        const int i0 = (t / nt) << 4, j0 = (t % nt) << 4;
        v8f c;
        if (kp == 0) {
          c = (v8f){0.f, 0.f, 0.f, 0.f, 0.f, 0.f, 0.f, 0.f};
        } else {
#pragma unroll
          for (int q = 0; q < 8; ++q)
            c[q] = A[(size_t)(i0 + q + half * 8) * stA + (j0 + r)];
        }
        const float* arow = WpT + (size_t)(i0 + r) * KPp + 2 * half;
        const float* brow = WpT + (size_t)(j0 + r) * KPp + 2 * half;
#pragma unroll 8
        for (int k4 = 0; k4 < KP; k4 += 4) {
          v2f a = *(const v2f*)(arow + k4);   // ds_load_b64, imm offset
          v2f b = *(const v2f*)(brow + k4);
          c = __builtin_amdgcn_wmma_f32_16x16x4_f32(false, a, false, b,
                                                    (short)0, c, false, false);
        }
        if (j0 + r < stA) {
#pragma unroll
          for (int q = 0; q < 8; ++q)
            A[(size_t)(i0 + q + half * 8) * stA + (j0 + r)] = c[q];
        }
      }
      __syncthreads();
    }
  }
  // UTLv (padded cols of U are zero)
  for (int i = tid; i < LDdim; i += 256) {
    float acc = 0.f;
    for (int m = 0; m < MIND; ++m) acc += U[(size_t)m * ldU + i] * pmv[m];
    v[i] = acc;
  }
  // + I / PRIOR_VAR (PRIOR_VAR = 1)
  for (int i = tid; i < n; i += 256) A[(size_t)i * stA + i] += 1.0f;
  __syncthreads();

  // ---- Cholesky #1: q_prec -> Lp (lower, in place) ----
  for (int k = 0; k < n; ++k) {
    if (tid == 0) A[(size_t)k * stA + k] = sqrtf(A[(size_t)k * stA + k]);
    __syncthreads();
    const float dk = A[(size_t)k * stA + k];
    for (int i = k + 1 + tid; i < n; i += 256) A[(size_t)i * stA + k] /= dk;
    __syncthreads();
    for (int j = k + 1 + tid; j < n; j += 256) {
      const float ajk = A[(size_t)j * stA + k];
      for (int i = j; i < n; ++i)
        A[(size_t)i * stA + j] -= A[(size_t)i * stA + k] * ajk;
    }
    __syncthreads();
  }

  // ---- invert Lp; Linv stored strictly-upper: A[j][i] = Linv[i][j] ----
  for (int i = tid; i < n; i += 256) dinv[i] = 1.f / A[(size_t)i * stA + i];
  __syncthreads();
  for (int j = tid; j < n; j += 256) {
    for (int i = j + 1; i < n; ++i) {
      float acc = A[(size_t)i * stA + j] * dinv[j];          // k = j term
      for (int k = j + 1; k < i; ++k)
        acc += A[(size_t)i * stA + k] * A[(size_t)j * stA + k];
      A[(size_t)j * stA + i] = -acc * dinv[i];
    }
  }
  __syncthreads();

  // tr(q_cov) = ||Linv||_F^2
  {
    float part = 0.f;
    for (int j = tid; j < n; j += 256) {
      part += dinv[j] * dinv[j];
      for (int i = j + 1; i < n; ++i) {
        const float x = A[(size_t)j * stA + i]; part += x * x;
      }
    }
    red[tid] = part; __syncthreads();
    for (int o = 128; o > 0; o >>= 1) {
      if (tid < o) red[tid] += red[tid + o];
      __syncthreads();
    }
  }
  const float tr_cov = red[0];
  __syncthreads();

  // q_mu = Linv^T (Linv v)
  for (int k = tid; k < n; k += 256) {
    float acc = dinv[k] * v[k];
    for (int j = 0; j < k; ++j) acc += A[(size_t)j * stA + k] * v[j];
    tv[k] = acc;
  }
  __syncthreads();
  for (int i = tid; i < n; i += 256) {
    float acc = dinv[i] * tv[i];
    for (int k = i + 1; k < n; ++k) acc += A[(size_t)i * stA + k] * tv[k];
    qmu[i] = acc;
  }
  __syncthreads();
  {
    float part = 0.f;
    for (int i = tid; i < n; i += 256) part += qmu[i] * qmu[i];
    red[tid] = part; __syncthreads();
    for (int o = 128; o > 0; o >>= 1) {
      if (tid < o) red[tid] += red[tid + o];
      __syncthreads();
    }
  }
  const float mahal = red[0];
  __syncthreads();

  // q_cov = Linv^T Linv into lower (reads upper + dinv only)
  for (int i = 1; i < n; ++i) {
    for (int j = tid; j < i; j += 256) {
      float acc = dinv[i] * A[(size_t)j * stA + i];
      for (int k = i + 1; k < n; ++k)
        acc += A[(size_t)i * stA + k] * A[(size_t)j * stA + k];
      A[(size_t)i * stA + j] = acc;
    }
  }
  __syncthreads();
  for (int i = tid; i < n; i += 256) {
    float acc = dinv[i] * dinv[i];
    for (int k = i + 1; k < n; ++k) {
      const float x = A[(size_t)i * stA + k]; acc += x * x;
    }
    A[(size_t)i * stA + i] = acc;
  }
  __syncthreads();

  // ---- Cholesky #2: q_cov -> Lc ----
  for (int k = 0; k < n; ++k) {
    if (tid == 0) A[(size_t)k * stA + k] = sqrtf(A[(size_t)k * stA + k]);
    __syncthreads();
    const float dk = A[(size_t)k * stA + k];
    for (int i = k + 1 + tid; i < n; i += 256) A[(size_t)i * stA + k] /= dk;
    __syncthreads();
    for (int j = k + 1 + tid; j < n; j += 256) {
      const float ajk = A[(size_t)j * stA + k];
      for (int i = j; i < n; ++i)
        A[(size_t)i * stA + j] -= A[(size_t)i * stA + k] * ajk;
    }
    __syncthreads();
  }
  {
    float part = 0.f;
    for (int i = tid; i < n; i += 256) part += logf(A[(size_t)i * stA + i]);
    red[tid] = part; __syncthreads();
    for (int o = 128; o > 0; o >>= 1) {
      if (tid < o) red[tid] += red[tid + o];
      __syncthreads();
    }
  }
  const float sumlc = red[0];   // logdet_cov = 2*sumlc
  __syncthreads();

  // w = q_mu + Lc eps ; padded row (stride 272, zeros past Din)
  const float* erow = eps + ((size_t)s * Dout + d) * n;
  float* wrow = wbuf + ((size_t)s * 256 + d) * 272;
  for (int i = tid; i < 272; i += 256) {
    float val = 0.f;
    if (i < n) {
      float acc = qmu[i];
      for (int j = 0; j <= i; ++j) acc += A[(size_t)i * stA + j] * erow[j];
      val = acc;
    }
    wrow[i] = val;
  }
  if (tid == 0)
    klpart[(size_t)s * 256 + d] =
        0.5f * (tr_cov + mahal - (float)n - 2.f * sumlc);
}

// ---------------------------------------------------------------------------
// Y[s,n,d] = act( sum_i X[s,n,i] * w[s,d,i] ) — one wave per 16x16 tile.
// ---------------------------------------------------------------------------
__global__ __launch_bounds__(32) void apply_wmma(
    const float* __restrict__ X, int ldIn,
    const float* __restrict__ wbuf,                 // [S, 256, 272]
    float* __restrict__ Y, int ldOut,
    int rows, int doElu)
{
  const int lane = threadIdx.x;
  const int r = lane & 15, half = lane >> 4;
  const int n0 = blockIdx.x << 4, d0 = blockIdx.y << 4, s = blockIdx.z;
  const float* xrow = X + ((size_t)s * rows + n0 + r) * ldIn;
  const float* wrow = wbuf + ((size_t)s * 256 + d0 + r) * 272;
  v8f c = (v8f){0.f, 0.f, 0.f, 0.f, 0.f, 0.f, 0.f, 0.f};
  for (int k = 0; k < ldIn; k += 4) {
    const int kk = k + 2 * half;                 // even -> 8B aligned
    v2f a = *(const v2f*)(xrow + kk);
    v2f b = *(const v2f*)(wrow + kk);
    c = __builtin_amdgcn_wmma_f32_16x16x4_f32(false, a, false, b,
                                              (short)0, c, false, false);
  }
  float* yb = Y + (size_t)s * rows * ldOut;
#pragma unroll
  for (int q = 0; q < 8; ++q) {
    float val = c[q];
    if (doElu) val = val > 0.f ? val : (expf(val) - 1.f);
    yb[(size_t)(n0 + q + half * 8) * ldOut + (d0 + r)] = val;
  }
}

// dst[s,row,i] = src[row,i] (i<cols), 1 at i==cols, 0 beyond (broadcast on S)
__global__ void pad_init(const float* __restrict__ src, float* __restrict__ dst,
                         int rows, int cols, int ldDst) {
  const size_t idx = (size_t)blockIdx.x * blockDim.x + threadIdx.x;
  const size_t total = (size_t)SAMP * rows * ldDst;
  if (idx >= total) return;
  const int i = (int)(idx % ldDst);
  const int row = (int)((idx / ldDst) % rows);
  float val;
  if (i < cols) val = src[(size_t)row * cols + i];
  else val = (i == cols) ? 1.f : 0.f;
  dst[idx] = val;
}

// fill cols [onesCol, ld): 1 at onesCol, 0 after (appends bias column)
__global__ void pad_tail(float* __restrict__ dst, int rows, int onesCol, int ld) {
  const int tail = ld - onesCol;
  const size_t idx = (size_t)blockIdx.x * blockDim.x + threadIdx.x;
  const size_t total = (size_t)SAMP * rows * tail;
  if (idx >= total) return;
  const int j = (int)(idx % tail);
  const size_t rn = idx / tail;
  dst[rn * ld + onesCol + j] = (j == 0) ? 1.f : 0.f;
}

__global__ void kl_reduce(const float* __restrict__ klparts,
                          float* __restrict__ out) {
  const int s = threadIdx.x;
  if (s >= SAMP) return;
  const int douts[3] = {256, 256, 64};
  float acc = 0.f;
  for (int l = 0; l < 3; ++l)
    for (int d = 0; d < douts[l]; ++d)
      acc += klparts[(size_t)l * SAMP * 256 + (size_t)s * 256 + d];
  out[s] = acc;
}

extern "C" void kernel_launch(void* const* d_in, const int* in_sizes, int n_in,
                              void* d_out, int out_size, void* d_ws,
                              size_t ws_size, hipStream_t stream) {
  (void)in_sizes; (void)n_in; (void)out_size; (void)ws_size;
  const float* F    = (const float*)d_in[0];
  const float* IP   = (const float*)d_in[1];
  const float* plp1 = (const float*)d_in[2];
  const float* pmu1 = (const float*)d_in[3];
  const float* plp2 = (const float*)d_in[4];
  const float* pmu2 = (const float*)d_in[5];
  const float* plp3 = (const float*)d_in[6];
  const float* pmu3 = (const float*)d_in[7];
  const float* eps1 = (const float*)d_in[8];
  const float* eps2 = (const float*)d_in[9];
  const float* eps3 = (const float*)d_in[10];
  float* out = (float*)d_out;

  float* ws = (float*)d_ws;
  const size_t fsz = (size_t)SAMP * NROW * 272;
  const size_t usz = (size_t)SAMP * MIND * 272;
  float* F0 = ws;
  float* F1 = F0 + fsz;
  float* U0 = F1 + fsz;
  float* U1 = U0 + usz;
  float* W  = U1 + usz;                      // [S,256,272]
  float* KL = W + (size_t)SAMP * 256 * 272;  // [3,S,256]

  auto shmem = [](int LDdim, int stA, int KP) {
    return (size_t)((size_t)LDdim * stA + (size_t)LDdim * (KP + 2) + 512 +
                    4 * (size_t)LDdim + 256) * sizeof(float);
  };
  const size_t sh1 = shmem(80, 80, 256);     // 112,512 B, single panel
  const size_t sh2 = shmem(272, 260, 32);    // 327,296 B <= 327,680 B LDS

  // initial bias-padded inputs (ld=80 for Din=65)
  {
    size_t tot = (size_t)SAMP * NROW * 80;
    pad_init<<<dim3((unsigned)((tot + 255) / 256)), 256, 0, stream>>>(F, F0, NROW, 64, 80);
    tot = (size_t)SAMP * MIND * 80;
    pad_init<<<dim3((unsigned)((tot + 255) / 256)), 256, 0, stream>>>(IP, U0, MIND, 64, 80);
  }

  // ---- Layer 1: Din=65 (LDdim=80, stA=80, KP=256), Dout=256, ELU ----
  gram_chol_sample<<<dim3(256, SAMP), 256, sh1, stream>>>(
      U0, 80, plp1, pmu1, eps1, W, KL, 65, 80, 80, 256, 256);
  apply_wmma<<<dim3(NROW / 16, 16, SAMP), 32, 0, stream>>>(F0, 80, W, F1, 272, NROW, 1);
  apply_wmma<<<dim3(MIND / 16, 16, SAMP), 32, 0, stream>>>(U0, 80, W, U1, 272, MIND, 1);
  pad_tail<<<dim3((unsigned)(((size_t)SAMP * NROW * 16 + 255) / 256)), 256, 0, stream>>>(F1, NROW, 256, 272);
  pad_tail<<<dim3((unsigned)(((size_t)SAMP * MIND * 16 + 255) / 256)), 256, 0, stream>>>(U1, MIND, 256, 272);

  // ---- Layer 2: Din=257 (LDdim=272, stA=260, KP=32), Dout=256, ELU ----
  gram_chol_sample<<<dim3(256, SAMP), 256, sh2, stream>>>(
      U1, 272, plp2, pmu2, eps2, W, KL + SAMP * 256, 257, 272, 260, 32, 256);
  apply_wmma<<<dim3(NROW / 16, 16, SAMP), 32, 0, stream>>>(F1, 272, W, F0, 272, NROW, 1);
  apply_wmma<<<dim3(MIND / 16, 16, SAMP), 32, 0, stream>>>(U1, 272, W, U0, 272, MIND, 1);
  pad_tail<<<dim3((unsigned)(((size_t)SAMP * NROW * 16 + 255) / 256)), 256, 0, stream>>>(F0, NROW, 256, 272);
  pad_tail<<<dim3((unsigned)(((size_t)SAMP * MIND * 16 + 255) / 256)), 256, 0, stream>>>(U0, MIND, 256, 272);

  // ---- Layer 3: Din=257, Dout=64, identity; Fs straight to d_out ----
  gram_chol_sample<<<dim3(64, SAMP), 256, sh2, stream>>>(
      U0, 272, plp3, pmu3, eps3, W, KL + 2 * SAMP * 256, 257, 272, 260, 32, 64);
  apply_wmma<<<dim3(NROW / 16, 4, SAMP), 32, 0, stream>>>(F0, 272, W, out, 64, NROW, 0);

  kl_reduce<<<dim3(1), 32, 0, stream>>>(KL, out + (size_t)SAMP * NROW * 64);
}